// Memory_5093831213770
// MI455X (gfx1250) — compile-verified
//
#include <hip/hip_runtime.h>
#include <math.h>

// Dims from the reference
#define A_DIM 256
#define H_DIM 1024
#define C_DIM 1024
#define N_DIM 8192
#define M_DIM 4096
#define X_DIM (A_DIM + H_DIM)   // 1280

typedef __attribute__((ext_vector_type(2))) float v2f;
typedef __attribute__((ext_vector_type(8))) float v8f;

__device__ __forceinline__ float wave_reduce_add(float v) {
    #pragma unroll
    for (int o = 16; o > 0; o >>= 1) v += __shfl_down(v, o, 32);
    return v;
}

// ---------------------------------------------------------------------------
// K1: GRU GEMVs. One wave32 per output row j of gi/gh (3072 rows).
//   gi[j] = dot(x, W_ih[j,:]) + b_ih[j],  x = [action | hidden_state]
//   gh[j] = dot(h_prev, W_hh[j,:]) + b_hh[j]
// float4 (b128) loads, wave shuffle reduction.
// ---------------------------------------------------------------------------
__global__ void __launch_bounds__(256)
gru_gemv(const float* __restrict__ action, const float* __restrict__ hidden,
         const float* __restrict__ hprev,
         const float* __restrict__ W_ih, const float* __restrict__ W_hh,
         const float* __restrict__ b_ih, const float* __restrict__ b_hh,
         float* __restrict__ gi, float* __restrict__ gh) {
    const int wave = (blockIdx.x * blockDim.x + threadIdx.x) >> 5;
    const int lane = threadIdx.x & 31;
    if (wave >= 3 * C_DIM) return;

    const float* wi = W_ih + (size_t)wave * X_DIM;
    const float* wh = W_hh + (size_t)wave * C_DIM;

    float s = 0.f;
    for (int k = lane * 4; k < X_DIM; k += 128) {
        float4 w = *(const float4*)(wi + k);
        float4 x = (k < A_DIM) ? *(const float4*)(action + k)
                               : *(const float4*)(hidden + (k - A_DIM));
        s += w.x * x.x + w.y * x.y + w.z * x.z + w.w * x.w;
    }
    float t = 0.f;
    for (int k = lane * 4; k < C_DIM; k += 128) {
        float4 w = *(const float4*)(wh + k);
        float4 h = *(const float4*)(hprev + k);
        t += w.x * h.x + w.y * h.y + w.z * h.z + w.w * h.w;
    }
    s = wave_reduce_add(s);
    t = wave_reduce_add(t);
    if (lane == 0) {
        gi[wave] = s + b_ih[wave];
        gh[wave] = t + b_hh[wave];
    }
}

// ---------------------------------------------------------------------------
// K2: GRU gate math -> h (1024 values), written to ws and to d_out tail.
// ---------------------------------------------------------------------------
__global__ void gru_combine(const float* __restrict__ gi, const float* __restrict__ gh,
                            const float* __restrict__ hprev,
                            float* __restrict__ h, float* __restrict__ h_out) {
    int c = blockIdx.x * blockDim.x + threadIdx.x;
    if (c >= C_DIM) return;
    float r  = 1.f / (1.f + expf(-(gi[c] + gh[c])));
    float z  = 1.f / (1.f + expf(-(gi[c + C_DIM] + gh[c + C_DIM])));
    float n  = tanhf(gi[c + 2 * C_DIM] + r * gh[c + 2 * C_DIM]);
    float hv = (1.f - z) * n + z * hprev[c];
    h[c] = hv;
    h_out[c] = hv;
}

// ---------------------------------------------------------------------------
// K3: scalar gate g = sigmoid(h.W_gate + b) and gamma = softplus(h.W_gamma + b).
// Also zeroes the read_vector accumulator region of d_out.
// ---------------------------------------------------------------------------
__global__ void __launch_bounds__(256)
scalars_kernel(const float* __restrict__ h,
               const float* __restrict__ W_gate, const float* __restrict__ b_gate,
               const float* __restrict__ W_gamma, const float* __restrict__ b_gamma,
               float* __restrict__ scal, float* __restrict__ read_vec_zero) {
    __shared__ float sa[8], sb[8];
    const int tid = threadIdx.x;
    float a = 0.f, b = 0.f;
    for (int k = tid; k < C_DIM; k += 256) {
        float hv = h[k];
        a += hv * W_gate[k];
        b += hv * W_gamma[k];
    }
    a = wave_reduce_add(a);
    b = wave_reduce_add(b);
    if ((tid & 31) == 0) { sa[tid >> 5] = a; sb[tid >> 5] = b; }
    __syncthreads();
    if (tid == 0) {
        float ta = 0.f, tb = 0.f;
        #pragma unroll
        for (int i = 0; i < 8; i++) { ta += sa[i]; tb += sb[i]; }
        float g = 1.f / (1.f + expf(-(ta + b_gate[0])));
        float x = tb + b_gamma[0];
        float gamma = (x > 20.f) ? x : log1pf(expf(x));   // softplus
        scal[0] = g;
        scal[1] = gamma;
    }
    for (int i = tid; i < M_DIM; i += 256) read_vec_zero[i] = 0.f;
}

// ---------------------------------------------------------------------------
// K4: interpolate + sharpen (circular-conv scalar cancels exactly under the
// sharpen normalization, sim-softmax over a size-1 axis is identically 1):
//   w = g*1 + (1-g)*prev ;  out = w^gamma / sum(w^gamma)
// Single 1024-thread block (tiny work), two-pass with block reduction.
// ---------------------------------------------------------------------------
__global__ void __launch_bounds__(1024)
weights_kernel(const float* __restrict__ prev_r, const float* __restrict__ prev_w,
               const float* __restrict__ scal,
               float* __restrict__ rw, float* __restrict__ ww) {
    __shared__ float srs[32], sws[32];
    __shared__ float totR, totW;
    const int tid = threadIdx.x;
    const float g = scal[0], gamma = scal[1], omg = 1.f - scal[0];

    float sr = 0.f, sw = 0.f;
    for (int n = tid; n < N_DIM; n += 1024) {
        sr += powf(g + omg * prev_r[n], gamma);
        sw += powf(g + omg * prev_w[n], gamma);
    }
    sr = wave_reduce_add(sr);
    sw = wave_reduce_add(sw);
    if ((tid & 31) == 0) { srs[tid >> 5] = sr; sws[tid >> 5] = sw; }
    __syncthreads();
    if (tid == 0) {
        float a = 0.f, b = 0.f;
        #pragma unroll
        for (int i = 0; i < 32; i++) { a += srs[i]; b += sws[i]; }
        totR = a; totW = b;
    }
    __syncthreads();
    const float ir = 1.f / totR, iw = 1.f / totW;
    for (int n = tid; n < N_DIM; n += 1024) {
        rw[n] = powf(g + omg * prev_r[n], gamma) * ir;
        ww[n] = powf(g + omg * prev_w[n], gamma) * iw;
    }
}

// ---------------------------------------------------------------------------
// K5 (hot pass, 134 MB): read_vector[m] = sum_n rw[n] * memory[n][m]
// via V_WMMA_F32_16X16X4_F32. Per wave:
//   A (16x4 f32): lane L holds memory[n + k][m0 + (L&15)] for
//                 k = {0,1} (lanes 0-15) / {2,3} (lanes 16-31)  -> v2f
//   B (4x16 f32): rw[n+k] broadcast across all 16 N-columns      -> v2f
//   D = A*B + C accumulates read_vector[m0..m0+15] replicated over columns.
// 2048 waves = 256 m-tiles x 8 N-slices; slices combined with f32 atomics.
// EXEC is all-ones through the WMMA loop (no divergence before writeback).
// ---------------------------------------------------------------------------
__global__ void __launch_bounds__(256)
read_vector_wmma(const float* __restrict__ mem, const float* __restrict__ rw,
                 float* __restrict__ out) {
    __shared__ float lds_rw[1024];
    const int tid  = threadIdx.x;
    const int wave = tid >> 5;
    const int lane = tid & 31;
    const int gw    = blockIdx.x * 8 + wave;
    const int slice = gw >> 8;        // uniform within a block
    const int mtile = gw & 255;
    const int nbase = slice * 1024;

    for (int i = tid; i < 1024; i += 256) lds_rw[i] = rw[nbase + i];
    __syncthreads();

    const int   half = lane >> 4;               // 0: K=0,1   1: K=2,3
    const int   mcol = (mtile << 4) + (lane & 15);
    const float* colp = mem + mcol;

    v8f c = {};
    #pragma unroll 4
    for (int t = 0; t < 1024; t += 4) {
        const int n = nbase + t + 2 * half;
        v2f a, b;
        a.x = colp[(size_t)n * M_DIM];
        a.y = colp[(size_t)(n + 1) * M_DIM];
        b.x = lds_rw[t + 2 * half];
        b.y = lds_rw[t + 2 * half + 1];
        c = __builtin_amdgcn_wmma_f32_16x16x4_f32(
                /*neg_a=*/false, a, /*neg_b=*/false, b,
                /*c_mod=*/(short)0, c, /*reuse_a=*/false, /*reuse_b=*/false);
    }

    // D columns are identical; lanes 0 and 16 own M rows 0-7 / 8-15.
    if ((lane & 15) == 0) {
        const int mbase = (mtile << 4) + (half << 3);
        #pragma unroll
        for (int v = 0; v < 8; v++) atomicAdd(&out[mbase + v], c[v]);
    }
}

// ---------------------------------------------------------------------------
extern "C" void kernel_launch(void* const* d_in, const int* in_sizes, int n_in,
                              void* d_out, int out_size, void* d_ws, size_t ws_size,
                              hipStream_t stream) {
    const float* action = (const float*)d_in[0];
    const float* hidden = (const float*)d_in[1];   // hidden_state (1,1,H)
    const float* prev_r = (const float*)d_in[2];
    const float* prev_w = (const float*)d_in[3];
    const float* hprev  = (const float*)d_in[4];   // prev_controller_hidden
    const float* memory = (const float*)d_in[5];
    const float* W_ih   = (const float*)d_in[6];
    const float* W_hh   = (const float*)d_in[7];
    const float* b_ih   = (const float*)d_in[8];
    const float* b_hh   = (const float*)d_in[9];
    const float* W_gamma = (const float*)d_in[20];
    const float* b_gamma = (const float*)d_in[21];
    const float* W_gate  = (const float*)d_in[22];
    const float* b_gate  = (const float*)d_in[23];

    float* out      = (float*)d_out;
    float* read_vec = out;                         // 4096
    float* rw_out   = out + 4096;                  // 8192
    float* ww_out   = out + 4096 + 8192;           // 8192
    float* h_out    = out + 4096 + 8192 + 8192;    // 1024

    float* ws   = (float*)d_ws;
    float* gi   = ws;            // 3072
    float* gh   = ws + 3072;     // 3072
    float* hbuf = ws + 6144;     // 1024
    float* scal = ws + 7168;     // 2: {g, gamma}

    gru_gemv<<<384, 256, 0, stream>>>(action, hidden, hprev, W_ih, W_hh,
                                      b_ih, b_hh, gi, gh);
    gru_combine<<<4, 256, 0, stream>>>(gi, gh, hprev, hbuf, h_out);
    scalars_kernel<<<1, 256, 0, stream>>>(hbuf, W_gate, b_gate, W_gamma, b_gamma,
                                          scal, read_vec);
    weights_kernel<<<1, 1024, 0, stream>>>(prev_r, prev_w, scal, rw_out, ww_out);
    read_vector_wmma<<<256, 256, 0, stream>>>(memory, rw_out, read_vec);
}